// PeakResponseLocalizer_31920196943984
// MI455X (gfx1250) — compile-verified
//
#include <hip/hip_runtime.h>

// ---------------------------------------------------------------------------
// Problem constants (from reference)
// ---------------------------------------------------------------------------
#define BB    8
#define CIN   4
#define HID   32
#define NCLS  19
#define HIN   512
#define WIN_  512
#define HO    256          // conv outputs are 256x256
#define WO    256
#define HU    512          // upsampled 512x512
#define WU    512
#define PLANE_UP (HU * WU)           // 262144
#define PLANE_SM (HO * WO)           // 65536
#define NBC   (BB * NCLS)            // 152
#define CMAPS_ELEMS ((size_t)BB * NCLS * HU * WU)   // 39,845,888
#define NBINS 1024
#define K_MED1 131071                 // 0-based lower middle of 262144
#define K_MED2 131072

// LDS row geometry: data starts at col 4 (16B aligned), left zero-pad at col 3,
// right zero-pad at col W+4. Row stride is a multiple of 4 floats.
#define RS1 520                       // conv1: 512 + pads, 16B-aligned stride
#define RS2 264                       // conv2: 256 + pads

typedef __attribute__((ext_vector_type(2))) float v2f;
typedef __attribute__((ext_vector_type(8))) float v8f;
typedef int v4i_ld __attribute__((vector_size(16)));   // matches builtin param

// Async global->LDS staging (CDNA5): guarded so the file compiles either way.
#ifndef HAVE_ASYNC_LDS
#if defined(__has_builtin)
#if __has_builtin(__builtin_amdgcn_global_load_async_to_lds_b128)
#define HAVE_ASYNC_LDS 1
#endif
#endif
#endif
#ifndef HAVE_ASYNC_LDS
#define HAVE_ASYNC_LDS 0
#endif

// Workspace layout (bytes)
#define OFF_H      ((size_t)0)                               // 67,108,864 B
#define OFF_CM     ((size_t)67108864)                        // 39,845,888 B
#define OFF_MINMAX ((size_t)106954752)                       // 2048 B
#define OFF_RANGE  ((size_t)106956800)                       // 2048 B (lo[256], hi[256])
#define OFF_HIST   ((size_t)106958848)                       // 622,592 B
#define OFF_CNTB   ((size_t)107581440)                       // 1024 B
#define OFF_MED    ((size_t)107582464)                       // 1024 B
#define OFF_SUMS   ((size_t)107583488)                       // 2048 B

// Order-preserving float <-> uint encoding for atomic min/max
static __device__ __forceinline__ unsigned enc_f32(float f) {
  unsigned u = __float_as_uint(f);
  return (u & 0x80000000u) ? ~u : (u | 0x80000000u);
}
static __device__ __forceinline__ float dec_f32(unsigned e) {
  unsigned u = (e & 0x80000000u) ? (e & 0x7fffffffu) : ~e;
  return __uint_as_float(u);
}

// k -> LDS word offset of (cin, kh, kw) within the staged padded image.
// Folds to a literal constant once the K-loop is fully unrolled.
static __device__ __forceinline__ constexpr int offA(int k, int rowstride) {
  return ((k / 9) * 3 + (k % 9) / 3) * rowstride + (k % 9) % 3;
}

// Stage one padded row (W data floats at col 4, zeros at cols 3 and W+4).
template <int W>
static __device__ __forceinline__ void stage_row(const float* __restrict__ g,
                                                 float* __restrict__ l,
                                                 bool valid, int tid) {
  if (valid) {
#if HAVE_ASYNC_LDS
    constexpr int NV = W / 4;                 // 16B per thread, 16B-aligned dst
    if (tid < NV)
      __builtin_amdgcn_global_load_async_to_lds_b128(
          (__attribute__((address_space(1))) v4i_ld*)(g + tid * 4),
          (__attribute__((address_space(3))) v4i_ld*)(l + 4 + tid * 4), 0, 0);
#else
    for (int c = tid; c < W; c += 256) l[4 + c] = g[c];
#endif
    if (tid == 0) { l[3] = 0.f; l[4 + W] = 0.f; }
  } else {
    for (int c = tid; c < W + 2; c += 256) l[3 + c] = 0.f;
  }
}

static __device__ __forceinline__ void stage_fence() {
#if HAVE_ASYNC_LDS
  asm volatile("s_wait_asynccnt 0x0" ::: "memory");
#endif
  __syncthreads();
}

// ---------------------------------------------------------------------------
// Kernel 1: conv1 (3x3, stride 2, pad 1) + bias + ReLU via WMMA f32 16x16x4.
// One block = one (b, oh) output row. Input rows staged zero-padded in LDS;
// weights repacked as (k-pair, n) float2 so B is a single ds_load_b64/step.
// ---------------------------------------------------------------------------
__global__ __launch_bounds__(256)
void conv1_relu_wmma(const float* __restrict__ x, const float* __restrict__ w1,
                     const float* __restrict__ b1, float* __restrict__ h) {
  __shared__ __align__(16) float xs[CIN * 3 * RS1];     // 24,960 B
  __shared__ __align__(16) float wp[36 * 32];           // pairs: [k/2][n][2]

  const int tid = threadIdx.x;
  const int b   = blockIdx.x >> 8;
  const int oh  = blockIdx.x & 255;
  const float* xb = x + (size_t)b * CIN * HIN * WIN_;

  // Stage weights as pairs: wp[((k>>1)*32 + n)*2 + (k&1)] = w1[n*36 + k]
  for (int i = tid; i < 36 * 32; i += 256) {
    int k = i >> 5, n = i & 31;
    wp[(((k >> 1) * 32 + n) << 1) | (k & 1)] = w1[n * 36 + k];
  }
  // Stage padded input rows (ih = 2*oh-1 .. 2*oh+1, all 4 cin)
#pragma unroll
  for (int cin = 0; cin < CIN; ++cin)
#pragma unroll
    for (int r = 0; r < 3; ++r) {
      int ih = 2 * oh - 1 + r;
      stage_row<WIN_>(xb + ((size_t)cin * HIN + ih) * WIN_,
                      xs + (cin * 3 + r) * RS1, (unsigned)ih < HIN, tid);
    }
  stage_fence();

  const int lane = tid & 31;
  const int wv   = tid >> 5;
  const int m    = lane & 15;
  const int nl   = lane & 15;
  const bool hi  = lane >= 16;        // lanes 16-31 hold K = k0+2, k0+3
  const v2f* wpv = (const v2f*)wp;
  const int wbase = (hi ? 32 : 0) + nl;   // + c0 added per tile

  // 32 tiles (16 ow-tiles x 2 channel-tiles) / 8 waves = 4 tiles per wave
#pragma unroll 1
  for (int tt = 0; tt < 4; ++tt) {
    const int tileid = wv * 4 + tt;
    const int ct  = tileid & 1;
    const int ow0 = (tileid >> 1) << 4;
    const int c0  = ct << 4;
    const int base_a = 2 * (ow0 + m) + 3;

    v8f acc = {};
#pragma unroll
    for (int j = 0; j < 9; ++j) {
      const int o0 = hi ? offA(4 * j + 2, RS1) : offA(4 * j + 0, RS1);
      const int o1 = hi ? offA(4 * j + 3, RS1) : offA(4 * j + 1, RS1);
      v2f a;
      a[0] = xs[base_a + o0];
      a[1] = xs[base_a + o1];
      v2f bv = wpv[wbase + c0 + 64 * j];
      acc = __builtin_amdgcn_wmma_f32_16x16x4_f32(
          false, a, false, bv, (short)0, acc, false, false);
    }

    const int c = c0 + nl;
    const float bias = b1[c];
    float* hp = h + (((size_t)b * HID + c) * HO + oh) * WO + ow0 + (hi ? 8 : 0);
#pragma unroll
    for (int r = 0; r < 8; ++r) {
      float v = acc[r] + bias;
      hp[r] = v > 0.f ? v : 0.f;
    }
  }
}

// ---------------------------------------------------------------------------
// Kernel 2: conv2 (3x3, stride 1, pad 1) + bias, K = 288 (72 WMMA steps).
// N padded 19 -> 32 (two 16-channel tiles; stores masked to c < 19).
// ---------------------------------------------------------------------------
__global__ __launch_bounds__(256)
void conv2_wmma(const float* __restrict__ h, const float* __restrict__ w2,
                const float* __restrict__ b2, float* __restrict__ cm) {
  __shared__ __align__(16) float xs[HID * 3 * RS2];     // 101,376 B
  __shared__ __align__(16) float wp[288 * 32];          // 36,864 B

  const int tid = threadIdx.x;
  const int b   = blockIdx.x >> 8;
  const int oh  = blockIdx.x & 255;
  const float* hb = h + (size_t)b * HID * HO * WO;

  for (int i = tid; i < 288 * 32; i += 256) {
    int k = i >> 5, n = i & 31;
    float w = (n < NCLS) ? w2[n * 288 + k] : 0.f;
    wp[(((k >> 1) * 32 + n) << 1) | (k & 1)] = w;
  }
#pragma unroll 4
  for (int cin = 0; cin < HID; ++cin)
#pragma unroll
    for (int r = 0; r < 3; ++r) {
      int ih = oh - 1 + r;
      stage_row<WO>(hb + ((size_t)cin * HO + ih) * WO,
                    xs + (cin * 3 + r) * RS2, (unsigned)ih < HO, tid);
    }
  stage_fence();

  const int lane = tid & 31;
  const int wv   = tid >> 5;
  const int m    = lane & 15;
  const int nl   = lane & 15;
  const bool hi  = lane >= 16;
  const v2f* wpv = (const v2f*)wp;
  const int wbase = (hi ? 32 : 0) + nl;

#pragma unroll 1
  for (int tt = 0; tt < 4; ++tt) {
    const int tileid = wv * 4 + tt;
    const int ct  = tileid & 1;
    const int ow0 = (tileid >> 1) << 4;
    const int c0  = ct << 4;
    const int base_a = ow0 + m + 3;

    v8f acc = {};
#pragma unroll
    for (int j = 0; j < 72; ++j) {
      const int o0 = hi ? offA(4 * j + 2, RS2) : offA(4 * j + 0, RS2);
      const int o1 = hi ? offA(4 * j + 3, RS2) : offA(4 * j + 1, RS2);
      v2f a;
      a[0] = xs[base_a + o0];
      a[1] = xs[base_a + o1];
      v2f bv = wpv[wbase + c0 + 64 * j];
      acc = __builtin_amdgcn_wmma_f32_16x16x4_f32(
          false, a, false, bv, (short)0, acc, false, false);
    }

    const int c = c0 + nl;
    if (c < NCLS) {
      const float bias = b2[c];
      float* cp = cm + (((size_t)b * NCLS + c) * HO + oh) * WO + ow0 + (hi ? 8 : 0);
#pragma unroll
      for (int r = 0; r < 8; ++r) cp[r] = acc[r] + bias;
    }
  }
}

// ---------------------------------------------------------------------------
// Kernel 3: bilinear 2x upsample fused with per-(b,c) min/max accumulation.
// ---------------------------------------------------------------------------
__global__ __launch_bounds__(256)
void upsample_minmax_kernel(const float* __restrict__ cm, float* __restrict__ out,
                            unsigned* __restrict__ minmax) {
  const int idx = blockIdx.x * 256 + threadIdx.x;
  const int bc  = idx >> 18;
  const int rem = idx & (PLANE_UP - 1);
  const int o   = rem >> 9;
  const int p   = rem & 511;

  const float s = 255.0f / 511.0f;          // (n_in-1)/(n_out-1)
  float ph = o * s, pw = p * s;
  int lh = (int)ph; lh = lh > 254 ? 254 : lh;
  int lw = (int)pw; lw = lw > 254 ? 254 : lw;
  float fh = ph - (float)lh, fw = pw - (float)lw;

  const float* plane = cm + (size_t)bc * PLANE_SM;
  const float* r0 = plane + (size_t)lh * WO + lw;
  float v00 = r0[0], v01 = r0[1], v10 = r0[WO], v11 = r0[WO + 1];
  float v = (1.f - fh) * ((1.f - fw) * v00 + fw * v01)
          +        fh  * ((1.f - fw) * v10 + fw * v11);
  out[(size_t)idx] = v;

  unsigned e = enc_f32(v);
  __shared__ unsigned lmin[256], lmax[256];
  lmin[threadIdx.x] = e;
  lmax[threadIdx.x] = e;
  __syncthreads();
  for (int st = 128; st > 0; st >>= 1) {
    if (threadIdx.x < st) {
      unsigned a0 = lmin[threadIdx.x], a1 = lmin[threadIdx.x + st];
      lmin[threadIdx.x] = a0 < a1 ? a0 : a1;
      unsigned b0 = lmax[threadIdx.x], b1v = lmax[threadIdx.x + st];
      lmax[threadIdx.x] = b0 > b1v ? b0 : b1v;
    }
    __syncthreads();
  }
  if (threadIdx.x == 0) {
    atomicMin(&minmax[2 * bc + 0], lmin[0]);
    atomicMax(&minmax[2 * bc + 1], lmax[0]);
  }
}

// ---------------------------------------------------------------------------
// Median machinery: init, range decode, histogram (x2), refine, final
// ---------------------------------------------------------------------------
__global__ void init_stats_kernel(unsigned* __restrict__ minmax,
                                  float* __restrict__ sums) {
  int i = blockIdx.x * 256 + threadIdx.x;
  if (i < NBC) {
    minmax[2 * i + 0] = 0xFFFFFFFFu;
    minmax[2 * i + 1] = 0u;
    sums[2 * i + 0] = 0.f;
    sums[2 * i + 1] = 0.f;
  }
}

__global__ void zero_hist_kernel(unsigned* __restrict__ hist) {
  int i = blockIdx.x * 256 + threadIdx.x;
  if (i < NBC * NBINS) hist[i] = 0u;
}

__global__ void range_from_minmax_kernel(const unsigned* __restrict__ minmax,
                                         float* __restrict__ lo_arr,
                                         float* __restrict__ hi_arr) {
  int i = threadIdx.x;
  if (i < NBC) {
    lo_arr[i] = dec_f32(minmax[2 * i + 0]);
    hi_arr[i] = dec_f32(minmax[2 * i + 1]);
  }
}

__global__ __launch_bounds__(256)
void hist_kernel(const float* __restrict__ src, const float* __restrict__ lo_arr,
                 const float* __restrict__ hi_arr, unsigned* __restrict__ hist) {
  __shared__ unsigned lh[NBINS];
  for (int i = threadIdx.x; i < NBINS; i += 256) lh[i] = 0u;
  __syncthreads();
  const int bc   = blockIdx.y;
  const float lo = lo_arr[bc], hi = hi_arr[bc];
  const float scale = (float)NBINS / fmaxf(hi - lo, 1e-30f);
  const size_t base = (size_t)bc * PLANE_UP + (size_t)blockIdx.x * 16384;
  for (int t = 0; t < 64; ++t) {
    float v = src[base + t * 256 + threadIdx.x];
    if (v >= lo && v <= hi) {
      int bin = (int)((v - lo) * scale);
      bin = bin < 0 ? 0 : (bin > NBINS - 1 ? NBINS - 1 : bin);
      atomicAdd(&lh[bin], 1u);
    }
  }
  __syncthreads();
  for (int i = threadIdx.x; i < NBINS; i += 256)
    if (lh[i]) atomicAdd(&hist[bc * NBINS + i], lh[i]);
}

__global__ void refine_kernel(const unsigned* __restrict__ hist,
                              float* __restrict__ lo_arr,
                              float* __restrict__ hi_arr,
                              unsigned* __restrict__ cnt_below) {
  int bc = threadIdx.x;
  if (bc >= NBC) return;
  const float lo = lo_arr[bc], hi = hi_arr[bc];
  const float binw = (hi - lo) / (float)NBINS;
  long cum = 0;
  int i1 = -1, i2 = -1;
  unsigned cb = 0;
  for (int i = 0; i < NBINS; ++i) {
    unsigned c = hist[bc * NBINS + i];
    if (i1 < 0 && cum + (long)c > (long)K_MED1) { i1 = i; cb = (unsigned)cum; }
    if (i2 < 0 && cum + (long)c > (long)K_MED2) { i2 = i; }
    cum += c;
  }
  if (i1 < 0) i1 = NBINS - 1;
  if (i2 < 0) i2 = NBINS - 1;
  lo_arr[bc] = lo + (float)i1 * binw;
  hi_arr[bc] = lo + (float)(i2 + 1) * binw;
  cnt_below[bc] = cb;
}

__global__ void median_final_kernel(const unsigned* __restrict__ hist,
                                    const float* __restrict__ lo_arr,
                                    const float* __restrict__ hi_arr,
                                    const unsigned* __restrict__ cnt_below,
                                    float* __restrict__ med) {
  int bc = threadIdx.x;
  if (bc >= NBC) return;
  const long k1 = (long)K_MED1 - (long)cnt_below[bc];
  const long k2 = (long)K_MED2 - (long)cnt_below[bc];
  const float lo = lo_arr[bc];
  const float binw = (hi_arr[bc] - lo) / (float)NBINS;
  long cum = 0;
  float m1 = lo, m2 = lo;
  bool f1 = false, f2 = false;
  for (int i = 0; i < NBINS; ++i) {
    unsigned c = hist[bc * NBINS + i];
    if (!f1 && cum + (long)c > k1) { m1 = lo + ((float)i + 0.5f) * binw; f1 = true; }
    if (!f2 && cum + (long)c > k2) { m2 = lo + ((float)i + 0.5f) * binw; f2 = true; }
    cum += c;
  }
  med[bc] = 0.5f * (m1 + m2);
}

// ---------------------------------------------------------------------------
// 3x3 local-max peak detection + masked sum/count reduction
// ---------------------------------------------------------------------------
__global__ __launch_bounds__(256)
void peak_kernel(const float* __restrict__ cmaps, const float* __restrict__ med,
                 float* __restrict__ sums) {
  const int idx = blockIdx.x * 256 + threadIdx.x;
  const int bc  = idx >> 18;
  const int rem = idx & (PLANE_UP - 1);
  const int o   = rem >> 9;
  const int p   = rem & 511;
  const float* plane = cmaps + (size_t)bc * PLANE_UP;
  const float v = plane[(size_t)o * WU + p];
  float mx = v;
#pragma unroll
  for (int dy = -1; dy <= 1; ++dy)
#pragma unroll
    for (int dx = -1; dx <= 1; ++dx) {
      int oo = o + dy, pp = p + dx;
      if (oo >= 0 && oo < HU && pp >= 0 && pp < WU)
        mx = fmaxf(mx, plane[(size_t)oo * WU + pp]);
    }
  const bool peak = (v == mx) && (v >= med[bc]);
  float s = peak ? v : 0.f;
  float c = peak ? 1.f : 0.f;

  __shared__ float ls[256], lc[256];
  ls[threadIdx.x] = s;
  lc[threadIdx.x] = c;
  __syncthreads();
  for (int st = 128; st > 0; st >>= 1) {
    if (threadIdx.x < st) {
      ls[threadIdx.x] += ls[threadIdx.x + st];
      lc[threadIdx.x] += lc[threadIdx.x + st];
    }
    __syncthreads();
  }
  if (threadIdx.x == 0) {
    atomicAdd(&sums[2 * bc + 0], ls[0]);
    atomicAdd(&sums[2 * bc + 1], lc[0]);
  }
}

__global__ void logits_kernel(const float* __restrict__ sums,
                              float* __restrict__ logits) {
  int i = threadIdx.x;
  if (i < NBC) logits[i] = sums[2 * i + 0] / sums[2 * i + 1];
}

// ---------------------------------------------------------------------------
// Launch
// ---------------------------------------------------------------------------
extern "C" void kernel_launch(void* const* d_in, const int* in_sizes, int n_in,
                              void* d_out, int out_size, void* d_ws, size_t ws_size,
                              hipStream_t stream) {
  const float* x  = (const float*)d_in[0];
  const float* w1 = (const float*)d_in[1];
  const float* b1 = (const float*)d_in[2];
  const float* w2 = (const float*)d_in[3];
  const float* b2 = (const float*)d_in[4];
  float* out = (float*)d_out;

  char* ws = (char*)d_ws;
  float*    h      = (float*)(ws + OFF_H);
  float*    cm     = (float*)(ws + OFF_CM);
  unsigned* minmax = (unsigned*)(ws + OFF_MINMAX);
  float*    lo_arr = (float*)(ws + OFF_RANGE);
  float*    hi_arr = lo_arr + 256;
  unsigned* hist   = (unsigned*)(ws + OFF_HIST);
  unsigned* cntb   = (unsigned*)(ws + OFF_CNTB);
  float*    med    = (float*)(ws + OFF_MED);
  float*    sums   = (float*)(ws + OFF_SUMS);
  float*    logits = out + CMAPS_ELEMS;

  const int conv_blocks = BB * HO;                        // 2048 (b, oh) rows
  const int up_blocks   = (int)(CMAPS_ELEMS / 256);       // 155648

  init_stats_kernel<<<1, 256, 0, stream>>>(minmax, sums);
  conv1_relu_wmma<<<conv_blocks, 256, 0, stream>>>(x, w1, b1, h);
  conv2_wmma<<<conv_blocks, 256, 0, stream>>>(h, w2, b2, cm);
  upsample_minmax_kernel<<<up_blocks, 256, 0, stream>>>(cm, out, minmax);
  range_from_minmax_kernel<<<1, 256, 0, stream>>>(minmax, lo_arr, hi_arr);

  zero_hist_kernel<<<(NBC * NBINS + 255) / 256, 256, 0, stream>>>(hist);
  dim3 hgrid(16, NBC);
  hist_kernel<<<hgrid, 256, 0, stream>>>(out, lo_arr, hi_arr, hist);
  refine_kernel<<<1, 256, 0, stream>>>(hist, lo_arr, hi_arr, cntb);
  zero_hist_kernel<<<(NBC * NBINS + 255) / 256, 256, 0, stream>>>(hist);
  hist_kernel<<<hgrid, 256, 0, stream>>>(out, lo_arr, hi_arr, hist);
  median_final_kernel<<<1, 256, 0, stream>>>(hist, lo_arr, hi_arr, cntb, med);

  peak_kernel<<<up_blocks, 256, 0, stream>>>(out, med, sums);
  logits_kernel<<<1, 256, 0, stream>>>(sums, logits);
}